// LATTE_64690797413065
// MI455X (gfx1250) — compile-verified
//
#include <hip/hip_runtime.h>
#include <cstdint>
#include <cstddef>

// Problem constants (match reference)
#define N_NODES 50000
#define F_DIM   256
#define D_DIM   128
#define E_EDGES 800000

typedef float v2f __attribute__((ext_vector_type(2)));
typedef float v8f __attribute__((ext_vector_type(8)));

// ---------------------------------------------------------------------------
// Kernel 1: h = x @ lin_W  via V_WMMA_F32_16X16X4_F32
//   block = 256 threads (8 waves). Block tile: 64 rows x 128 cols.
//   wave w: row tile (w>>1)*16, column half (w&1)*64 -> 4 WMMA 16x16 tiles.
//   K staged in LDS panels of 64 (4 panels over F=256).
//
// B-operand optimization: W is staged into LDS as float2 K-row pairs
//   wpk[kp][n] = { W[2kp][n], W[2kp+1][n] }
// so each WMMA B fragment is ONE aligned ds_read_b64 into an even register
// pair (no v_mov repacking). WP=144 float2 stride => consecutive pair-rows
// are skewed by 32 banks, so lanes 0-15 (row kp0) and 16-31 (row kp0+1)
// hit disjoint bank halves: conflict-free.
// ---------------------------------------------------------------------------
#define KP  64
#define XS  66     // x-panel float row stride (pad keeps float2 alignment)
#define WP  144    // packed-W float2 row stride

__global__ __launch_bounds__(256) void gemm_h_kernel(const float* __restrict__ x,
                                                     const float* __restrict__ W,
                                                     float* __restrict__ h)
{
    __shared__ float  xs [64 * XS];          // 64 rows x 64 K (padded)      ~16.9 KB
    __shared__ float2 wpk[(KP / 2) * WP];    // 32 K-pairs x 128 N (padded)  ~36.9 KB

    const int tid  = threadIdx.x;
    const int wave = tid >> 5;
    const int lane = tid & 31;
    const int half = lane >> 4;      // 0: lanes 0-15, 1: lanes 16-31
    const int mi   = lane & 15;      // M (A) / N (B,C) index within tile
    const int m0   = blockIdx.x * 64 + (wave >> 1) * 16;
    const int nb   = (wave & 1) * 64;

    v8f acc[4];
#pragma unroll
    for (int t = 0; t < 4; ++t) acc[t] = (v8f){0.f,0.f,0.f,0.f,0.f,0.f,0.f,0.f};

    for (int p = 0; p < F_DIM / KP; ++p) {
        const int kb = p * KP;
        __syncthreads();
        {   // stage x panel: 64 rows x 64 cols, 16 floats/thread
            const int r    = tid >> 2;
            const int cb   = (tid & 3) * 16;
            const int grow = blockIdx.x * 64 + r;
            if (grow < N_NODES) {
                const float* src = x + (size_t)grow * F_DIM + kb + cb;
                float*       dst = xs + r * XS + cb;
#pragma unroll
                for (int j = 0; j < 16; j += 4)
                    *(float4*)(dst + j) = *(const float4*)(src + j);
            }
        }
        {   // stage W panel pre-paired: thread t -> K-pair kp, 16 columns
            const int kp = tid >> 3;            // 0..31
            const int cb = (tid & 7) * 16;      // 0..112
            const float* s0 = W + (size_t)(kb + 2 * kp)     * D_DIM + cb;
            const float* s1 = W + (size_t)(kb + 2 * kp + 1) * D_DIM + cb;
            float2* dst = wpk + kp * WP + cb;
#pragma unroll
            for (int j = 0; j < 16; j += 4) {
                float4 r0 = *(const float4*)(s0 + j);
                float4 r1 = *(const float4*)(s1 + j);
                dst[j + 0] = make_float2(r0.x, r1.x);
                dst[j + 1] = make_float2(r0.y, r1.y);
                dst[j + 2] = make_float2(r0.z, r1.z);
                dst[j + 3] = make_float2(r0.w, r1.w);
            }
        }
        __syncthreads();

        if (m0 < N_NODES) {          // wave-uniform: EXEC all-1s inside (WMMA req.)
#pragma unroll 4
            for (int kk = 0; kk < KP; kk += 4) {
                // A 16x4 f32 layout: lanes0-15 K={kk,kk+1}, lanes16-31 K={kk+2,kk+3}
                v2f a = *(const v2f*)(xs + mi * XS + kk + 2 * half);
                // B pair row: kp0 = kk/2 (+1 for upper lane half)
                const float2* brow = wpk + ((kk >> 1) + half) * WP + nb + mi;
#pragma unroll
                for (int t = 0; t < 4; ++t) {
                    v2f b = *(const v2f*)(brow + t * 16);   // one ds_read_b64
                    acc[t] = __builtin_amdgcn_wmma_f32_16x16x4_f32(
                        false, a, false, b, (short)0, acc[t], false, false);
                }
            }
        }
    }

    if (m0 < N_NODES) {
        // C/D layout: row = m0 + half*8 + v, col = nb + t*16 + mi
#pragma unroll
        for (int t = 0; t < 4; ++t)
#pragma unroll
            for (int v = 0; v < 8; ++v)
                h[(size_t)(m0 + half * 8 + v) * D_DIM + nb + t * 16 + mi] = acc[t][v];
    }
}

// ---------------------------------------------------------------------------
// Kernel 2: per-node fused  beta = softmax(x@convW^T + b),
//           score_l = h@attn_l_w^T + b,  score_r = h@attn_r_w^T + b.
// One wave (32 lanes) per node; shuffle reduction (wave32).
// ---------------------------------------------------------------------------
__device__ __forceinline__ float wave_sum(float v) {
#pragma unroll
    for (int m = 16; m > 0; m >>= 1) v += __shfl_xor(v, m, 32);
    return v;
}

__global__ __launch_bounds__(256) void scores_beta_kernel(
    const float* __restrict__ x,   const float* __restrict__ h,
    const float* __restrict__ convW, const float* __restrict__ convB,
    const float* __restrict__ alw, const float* __restrict__ alb,
    const float* __restrict__ arw, const float* __restrict__ arb,
    float* __restrict__ beta, float* __restrict__ sl, float* __restrict__ sr)
{
    const int node = (int)((blockIdx.x * blockDim.x + threadIdx.x) >> 5);
    const int lane = threadIdx.x & 31;
    if (node >= N_NODES) return;

    // conv logits: each lane covers 8 of F=256 columns
    const float* xr = x + (size_t)node * F_DIM + lane * 8;
    float c0 = 0.f, c1 = 0.f, c2 = 0.f;
#pragma unroll
    for (int j = 0; j < 8; j += 4) {
        float4 xv = *(const float4*)(xr + j);
        float4 w0 = *(const float4*)(convW + 0 * F_DIM + lane * 8 + j);
        float4 w1 = *(const float4*)(convW + 1 * F_DIM + lane * 8 + j);
        float4 w2 = *(const float4*)(convW + 2 * F_DIM + lane * 8 + j);
        c0 += xv.x*w0.x + xv.y*w0.y + xv.z*w0.z + xv.w*w0.w;
        c1 += xv.x*w1.x + xv.y*w1.y + xv.z*w1.z + xv.w*w1.w;
        c2 += xv.x*w2.x + xv.y*w2.y + xv.z*w2.z + xv.w*w2.w;
    }
    // attention scores: each lane covers 4 of D=128 columns
    float4 hv  = *(const float4*)(h   + (size_t)node * D_DIM + lane * 4);
    float4 wl0 = *(const float4*)(alw + 0 * D_DIM + lane * 4);
    float4 wl1 = *(const float4*)(alw + 1 * D_DIM + lane * 4);
    float4 wr0 = *(const float4*)(arw + 0 * D_DIM + lane * 4);
    float4 wr1 = *(const float4*)(arw + 1 * D_DIM + lane * 4);
    float l0 = hv.x*wl0.x + hv.y*wl0.y + hv.z*wl0.z + hv.w*wl0.w;
    float l1 = hv.x*wl1.x + hv.y*wl1.y + hv.z*wl1.z + hv.w*wl1.w;
    float r0 = hv.x*wr0.x + hv.y*wr0.y + hv.z*wr0.z + hv.w*wr0.w;
    float r1 = hv.x*wr1.x + hv.y*wr1.y + hv.z*wr1.z + hv.w*wr1.w;

    c0 = wave_sum(c0); c1 = wave_sum(c1); c2 = wave_sum(c2);
    l0 = wave_sum(l0); l1 = wave_sum(l1);
    r0 = wave_sum(r0); r1 = wave_sum(r1);

    if (lane == 0) {
        c0 += convB[0]; c1 += convB[1]; c2 += convB[2];
        float mx = fmaxf(c0, fmaxf(c1, c2));
        float e0 = __expf(c0 - mx), e1 = __expf(c1 - mx), e2 = __expf(c2 - mx);
        float inv = 1.f / (e0 + e1 + e2);
        beta[node * 3 + 0] = e0 * inv;
        beta[node * 3 + 1] = e1 * inv;
        beta[node * 3 + 2] = e2 * inv;
        sl[node * 2 + 0] = l0 + alb[0];
        sl[node * 2 + 1] = l1 + alb[1];
        sr[node * 2 + 0] = r0 + arb[0];
        sr[node * 2 + 1] = r1 + arb[1];
    }
}

// ---------------------------------------------------------------------------
// Segment softmax, 3 passes. Float max via order-preserving int map so
// atomic max is a single native integer atomic (no CAS loop).
// ---------------------------------------------------------------------------
__device__ __forceinline__ int   f2ord(float f) { int i = __float_as_int(f); return i >= 0 ? i : (i ^ 0x7fffffff); }
__device__ __forceinline__ float ord2f(int i)   { return __int_as_float(i >= 0 ? i : (i ^ 0x7fffffff)); }

__global__ __launch_bounds__(256) void init_ms_kernel(int* m1, float* s1, int* m2, float* s2)
{
    const int i = blockIdx.x * blockDim.x + threadIdx.x;
    if (i < N_NODES) {
        const int neg = f2ord(-__builtin_inff());
        m1[i] = neg; m2[i] = neg;
        s1[i] = 0.f; s2[i] = 0.f;
    }
}

// out = beta[:,2] * h  (self relation); aggregations accumulate on top.
__global__ __launch_bounds__(256) void init_out_kernel(const float* __restrict__ h,
                                                       const float* __restrict__ beta,
                                                       float* __restrict__ out)
{
    const int i = blockIdx.x * blockDim.x + threadIdx.x;   // over N*D/4 float4
    if (i < N_NODES * (D_DIM / 4)) {
        const int   node = i >> 5;                         // D/4 == 32
        const float b2   = beta[node * 3 + 2];
        float4 hv = ((const float4*)h)[i];
        float4 o;  o.x = hv.x * b2; o.y = hv.y * b2; o.z = hv.z * b2; o.w = hv.w * b2;
        ((float4*)out)[i] = o;
    }
}

// Pass 1: alpha = sr[tail] + sl[head]; segment max via atomic int-max.
__global__ __launch_bounds__(256) void edge_alpha_max_kernel(
    const int* __restrict__ ei, const float* __restrict__ sl, const float* __restrict__ sr,
    const int r, float* __restrict__ alpha, int* __restrict__ mord)
{
    const int e = blockIdx.x * blockDim.x + threadIdx.x;
    if (e >= E_EDGES) return;
    const int head = ei[e];
    const int tail = ei[E_EDGES + e];
    const float a = sr[tail * 2 + r] + sl[head * 2 + r];
    alpha[e] = a;
    __hip_atomic_fetch_max(&mord[head], f2ord(a), __ATOMIC_RELAXED, __HIP_MEMORY_SCOPE_AGENT);
}

// Pass 2: e = exp(alpha - m[head]); s[head] += e  (native global_atomic_add_f32).
__global__ __launch_bounds__(256) void edge_exp_sum_kernel(
    const int* __restrict__ ei, const int* __restrict__ mord,
    float* __restrict__ alpha, float* __restrict__ s)
{
    const int e = blockIdx.x * blockDim.x + threadIdx.x;
    if (e >= E_EDGES) return;
    const int head = ei[e];
    const float m  = ord2f(mord[head]);
    const float ee = __expf(alpha[e] - m);
    alpha[e] = ee;
    __hip_atomic_fetch_add(&s[head], ee, __ATOMIC_RELAXED, __HIP_MEMORY_SCOPE_AGENT);
}

// Pass 3: wave-per-edge; out[head] += (e/s[head]) * beta_r[head] * h[tail].
// h and out are L2-resident (25.6 MB each << 192 MB), atomics resolve in L2.
__global__ __launch_bounds__(256) void edge_scatter_kernel(
    const int* __restrict__ ei, const float* __restrict__ alpha,
    const float* __restrict__ s, const float* __restrict__ beta, const int r,
    const float* __restrict__ h, float* __restrict__ out)
{
    const int e    = (int)((blockIdx.x * blockDim.x + threadIdx.x) >> 5);
    const int lane = threadIdx.x & 31;
    if (e >= E_EDGES) return;

    if (lane == 0 && e + 8192 < E_EDGES) {     // stream-ahead on index arrays
        __builtin_prefetch(&ei[e + 8192], 0, 0);
        __builtin_prefetch(&ei[E_EDGES + e + 8192], 0, 0);
    }

    const int   head = ei[e];
    const int   tail = ei[E_EDGES + e];
    const float w = alpha[e] / (s[head] + 1e-16f) * beta[head * 3 + r];

    const float4 hv = *(const float4*)(h + (size_t)tail * D_DIM + lane * 4);
    float* o = out + (size_t)head * D_DIM + lane * 4;
    __hip_atomic_fetch_add(o + 0, hv.x * w, __ATOMIC_RELAXED, __HIP_MEMORY_SCOPE_AGENT);
    __hip_atomic_fetch_add(o + 1, hv.y * w, __ATOMIC_RELAXED, __HIP_MEMORY_SCOPE_AGENT);
    __hip_atomic_fetch_add(o + 2, hv.z * w, __ATOMIC_RELAXED, __HIP_MEMORY_SCOPE_AGENT);
    __hip_atomic_fetch_add(o + 3, hv.w * w, __ATOMIC_RELAXED, __HIP_MEMORY_SCOPE_AGENT);
}

// ---------------------------------------------------------------------------
// Host launcher
// ---------------------------------------------------------------------------
extern "C" void kernel_launch(void* const* d_in, const int* in_sizes, int n_in,
                              void* d_out, int out_size, void* d_ws, size_t ws_size,
                              hipStream_t stream)
{
    (void)in_sizes; (void)n_in; (void)out_size; (void)ws_size;

    const float* x     = (const float*)d_in[0];
    // d_in[1] = x_index (unused)
    const int*   ei1   = (const int*)d_in[2];
    const int*   ei2   = (const int*)d_in[3];
    const float* linW  = (const float*)d_in[4];
    const float* convW = (const float*)d_in[5];
    const float* convB = (const float*)d_in[6];
    const float* alw   = (const float*)d_in[7];
    const float* alb   = (const float*)d_in[8];
    const float* arw   = (const float*)d_in[9];
    const float* arb   = (const float*)d_in[10];
    float* out = (float*)d_out;

    // workspace carve-up (floats)
    float* w      = (float*)d_ws;
    float* h      = w;                       w += (size_t)N_NODES * D_DIM;   // 6.4M
    float* sl     = w;                       w += (size_t)N_NODES * 2;
    float* sr     = w;                       w += (size_t)N_NODES * 2;
    float* beta   = w;                       w += (size_t)N_NODES * 3;
    int*   m1     = (int*)w;                 w += (size_t)N_NODES;
    float* s1     = w;                       w += (size_t)N_NODES;
    int*   m2     = (int*)w;                 w += (size_t)N_NODES;
    float* s2     = w;                       w += (size_t)N_NODES;
    float* a1     = w;                       w += (size_t)E_EDGES;
    float* a2     = w;                       w += (size_t)E_EDGES;

    const int gemm_blocks  = (N_NODES + 63) / 64;                 // 782
    const int score_blocks = (N_NODES * 32 + 255) / 256;          // 6250
    const int ms_blocks    = (N_NODES + 255) / 256;               // 196
    const int io_blocks    = (N_NODES * (D_DIM / 4) + 255) / 256; // 6250
    const int e_blocks     = (E_EDGES + 255) / 256;               // 3125
    const int sc_blocks    = (E_EDGES * 32 + 255) / 256;          // 100000

    gemm_h_kernel<<<gemm_blocks, 256, 0, stream>>>(x, linW, h);
    scores_beta_kernel<<<score_blocks, 256, 0, stream>>>(x, h, convW, convB,
                                                         alw, alb, arw, arb,
                                                         beta, sl, sr);
    init_ms_kernel<<<ms_blocks, 256, 0, stream>>>(m1, s1, m2, s2);
    init_out_kernel<<<io_blocks, 256, 0, stream>>>(h, beta, out);

    // metapath 1 (relation 0)
    edge_alpha_max_kernel<<<e_blocks, 256, 0, stream>>>(ei1, sl, sr, 0, a1, m1);
    edge_exp_sum_kernel<<<e_blocks, 256, 0, stream>>>(ei1, m1, a1, s1);
    edge_scatter_kernel<<<sc_blocks, 256, 0, stream>>>(ei1, a1, s1, beta, 0, h, out);

    // metapath 2 (relation 1)
    edge_alpha_max_kernel<<<e_blocks, 256, 0, stream>>>(ei2, sl, sr, 1, a2, m2);
    edge_exp_sum_kernel<<<e_blocks, 256, 0, stream>>>(ei2, m2, a2, s2);
    edge_scatter_kernel<<<sc_blocks, 256, 0, stream>>>(ei2, a2, s2, beta, 1, h, out);
}